// Model_88957362634942
// MI455X (gfx1250) — compile-verified
//
#include <hip/hip_runtime.h>
#include <hip/hip_bf16.h>
#include <math.h>

#define J 34
#define NV 500000
#define NB 32
#define NTILES (NV / 16)   // 31250, exact

typedef float v2f __attribute__((ext_vector_type(2)));
typedef float v8f __attribute__((ext_vector_type(8)));

__device__ __constant__ int PARENTS_D[J] = {
    -1, 0, 1, 2, 2, 4, 5, 6, 7, 8, 9, 7, 11, 12, 7, 14, 15, 2,
    17, 18, 19, 20, 21, 22, 20, 24, 25, 20, 27, 28, 0, 30, 0, 32
};

// ---------------------------------------------------------------------------
// Setup: Rodrigues + forward kinematics -> zero-padded AfkT[16][36] in d_ws.
// AfkT[m][k] = A_k[m/4][m%4] for m<12, k<34; 0 otherwise.
// ---------------------------------------------------------------------------
__global__ void fk_setup(const float* __restrict__ pose,
                         const float* __restrict__ joints,
                         float* __restrict__ afkT) {
    __shared__ float Tl[J][16];   // local 4x4 transforms
    __shared__ float G[J][16];    // composed transforms
    const int tid = threadIdx.x;

    if (tid < J) {
        float rx = pose[tid * 3 + 0];
        float ry = pose[tid * 3 + 1];
        float rz = pose[tid * 3 + 2];
        float ang = sqrtf(rx * rx + ry * ry + rz * rz + 1e-8f);
        float inv = 1.0f / ang;
        float x = rx * inv, y = ry * inv, z = rz * inv;
        float s = sinf(ang), c = cosf(ang), oc = 1.0f - c;
        // K = [[0,-z,y],[z,0,-x],[-y,x,0]], K2 = K*K
        float R00 = 1.0f + oc * (-z * z - y * y);
        float R01 = -s * z + oc * (x * y);
        float R02 =  s * y + oc * (x * z);
        float R10 =  s * z + oc * (x * y);
        float R11 = 1.0f + oc * (-z * z - x * x);
        float R12 = -s * x + oc * (y * z);
        float R20 = -s * y + oc * (x * z);
        float R21 =  s * x + oc * (y * z);
        float R22 = 1.0f + oc * (-y * y - x * x);

        int p = PARENTS_D[tid];
        float tx = joints[tid * 3 + 0];
        float ty = joints[tid * 3 + 1];
        float tz = joints[tid * 3 + 2];
        if (p >= 0) {
            tx -= joints[p * 3 + 0];
            ty -= joints[p * 3 + 1];
            tz -= joints[p * 3 + 2];
        }
        Tl[tid][0] = R00;  Tl[tid][1] = R01;  Tl[tid][2]  = R02;  Tl[tid][3]  = tx;
        Tl[tid][4] = R10;  Tl[tid][5] = R11;  Tl[tid][6]  = R12;  Tl[tid][7]  = ty;
        Tl[tid][8] = R20;  Tl[tid][9] = R21;  Tl[tid][10] = R22;  Tl[tid][11] = tz;
        Tl[tid][12] = 0.0f; Tl[tid][13] = 0.0f; Tl[tid][14] = 0.0f; Tl[tid][15] = 1.0f;
    }
    __syncthreads();

    if (tid == 0) {
        for (int q = 0; q < 16; ++q) G[0][q] = Tl[0][q];
        for (int i = 1; i < J; ++i) {
            int p = PARENTS_D[i];
            float res[16];
            for (int r = 0; r < 4; ++r)
                for (int cc = 0; cc < 4; ++cc) {
                    float acc = 0.0f;
                    for (int k = 0; k < 4; ++k)
                        acc += G[p][r * 4 + k] * Tl[i][k * 4 + cc];
                    res[r * 4 + cc] = acc;
                }
            for (int q = 0; q < 16; ++q) G[i][q] = res[q];
        }
    }
    __syncthreads();

    // Emit transposed, padded Afk: 16 rows (m) x 36 cols (k=joint)
    for (int idx = tid; idx < 16 * 36; idx += blockDim.x) {
        int m = idx / 36;
        int k = idx % 36;
        float val = 0.0f;
        if (m < 12 && k < J) {
            int a = m >> 2, b = m & 3;
            if (b < 3) {
                val = G[k][a * 4 + b];
            } else {
                float jx = joints[k * 3 + 0];
                float jy = joints[k * 3 + 1];
                float jz = joints[k * 3 + 2];
                val = G[k][a * 4 + 3] -
                      (G[k][a * 4 + 0] * jx + G[k][a * 4 + 1] * jy + G[k][a * 4 + 2] * jz);
            }
        }
        afkT[idx] = val;
    }
}

// ---------------------------------------------------------------------------
// Main fused LBS kernel. One wave per 16-vertex tile.
// D(16x16) = AfkT(16x36, padded) x W^T(36x16)  via 9x V_WMMA_F32_16X16X4_F32.
// D column n = vertex (tile*16+n); rows 0..11 are its flattened 3x4 T matrix.
// ---------------------------------------------------------------------------
__global__ void __launch_bounds__(256)
lbs_wmma(const float* __restrict__ weights,      // V x 34
         const float* __restrict__ v_template,   // V x 3
         const float* __restrict__ local_adjust, // V x 3
         const float* __restrict__ displacement, // 3
         const float* __restrict__ scale,        // 1
         const float* __restrict__ est,          // B x 3
         const float* __restrict__ afkT,         // 16 x 36 (padded)
         float* __restrict__ out) {              // B x V x 3
    const int wave = threadIdx.x >> 5;
    const int lane = threadIdx.x & 31;
    const int tile = blockIdx.x * (blockDim.x >> 5) + wave;
    if (tile >= NTILES) return;                   // wave-uniform exit

    const int n    = lane & 15;                   // vertex within tile / matrix row
    const int hi   = lane >> 4;                   // half-wave select
    const int koff = hi * 2;                      // K sub-offset inside 4-wide tile
    const int v    = tile * 16 + n;

    const float* wrow = weights + (size_t)v * J;

    v8f acc = {0.f, 0.f, 0.f, 0.f, 0.f, 0.f, 0.f, 0.f};
#pragma unroll
    for (int t = 0; t < 9; ++t) {
        const int k0 = 4 * t;
        // A operand: rows of AfkT (zero-padded -> exact despite W pad garbage)
        v2f a;
        a.x = afkT[n * 36 + k0 + koff];
        a.y = afkT[n * 36 + k0 + koff + 1];
        // B operand: W^T chunk, per-lane 8-byte aligned load of 2 weights
        v2f b;
        if (t == 8 && hi) {
            b.x = 0.0f; b.y = 0.0f;               // k=34,35 pad (avoid OOB)
        } else {
            b = *(const v2f*)(wrow + k0 + koff);
        }
        acc = __builtin_amdgcn_wmma_f32_16x16x4_f32(
            /*neg_a=*/false, a, /*neg_b=*/false, b,
            /*c_mod=*/(short)0, acc, /*reuse_a=*/false, /*reuse_b=*/false);
    }

    // Per-lane posed vertex components.
    const float vx = v_template[v * 3 + 0] + local_adjust[v * 3 + 0];
    const float vy = v_template[v * 3 + 1] + local_adjust[v * 3 + 1];
    const float vz = v_template[v * 3 + 2] + local_adjust[v * 3 + 2];

    // lanes 0-15 hold T rows 0,1 (acc[0..7]); lanes 16-31 hold T row 2 (acc[0..3]).
    float p0 = acc[0] * vx + acc[1] * vy + acc[2] * vz + acc[3];
    float p1 = hi ? 0.0f : (acc[4] * vx + acc[5] * vy + acc[6] * vz + acc[7]);

    // Exchange across half-waves so every lane has (x,y,z).
    float q0 = __shfl_xor(p0, 16, 32);  // lower gets z, upper gets x
    float q1 = __shfl_xor(p1, 16, 32);  // upper gets y
    float x0, x1, x2;
    if (!hi) { x0 = p0; x1 = p1; x2 = q0; }
    else     { x0 = q0; x1 = q1; x2 = p0; }

    const float s  = scale[0];
    x0 = s * x0 + displacement[0];
    x1 = s * x1 + displacement[1];
    x2 = s * x2 + displacement[2];

    // Broadcast to 16 batches per half-wave (32 total), coalesced stores.
    const size_t vv = (size_t)v;
#pragma unroll
    for (int i = 0; i < 16; ++i) {
        const int b = hi * 16 + i;
        const float ex = est[b * 3 + 0];
        const float ey = est[b * 3 + 1];
        const float ez = est[b * 3 + 2];
        const size_t o = ((size_t)b * NV + vv) * 3;
        out[o + 0] = x0 + ex;
        out[o + 1] = x1 + ey;
        out[o + 2] = x2 + ez;
    }
}

// ---------------------------------------------------------------------------
extern "C" void kernel_launch(void* const* d_in, const int* in_sizes, int n_in,
                              void* d_out, int out_size, void* d_ws, size_t ws_size,
                              hipStream_t stream) {
    const float* pose         = (const float*)d_in[0];  // (34,3)
    const float* joints       = (const float*)d_in[1];  // (34,3)
    const float* weights      = (const float*)d_in[2];  // (V,34)
    const float* v_template   = (const float*)d_in[3];  // (V,3)
    const float* local_adjust = (const float*)d_in[4];  // (V,3)
    const float* displacement = (const float*)d_in[5];  // (1,3)
    const float* scale        = (const float*)d_in[6];  // (1,)
    const float* est          = (const float*)d_in[7];  // (B,3)

    float* afkT = (float*)d_ws;                          // 16*36 floats

    fk_setup<<<1, 64, 0, stream>>>(pose, joints, afkT);

    const int wavesPerBlock = 8;                         // 256 threads
    const int blocks = (NTILES + wavesPerBlock - 1) / wavesPerBlock;
    lbs_wmma<<<blocks, 256, 0, stream>>>(weights, v_template, local_adjust,
                                         displacement, scale, est, afkT,
                                         (float*)d_out);
}